// ClueCausalityExtractionThesis_60438779789340
// MI455X (gfx1250) — compile-verified
//
#include <hip/hip_runtime.h>
#include <math.h>

// Problem dims
#define BB 16
#define TT 1024
#define KK 8
#define DD 768
#define HH 384
#define TH3 1152     // 3*H
#define TP1 1025     // T+1

typedef __attribute__((ext_vector_type(2))) float v2f;
typedef __attribute__((ext_vector_type(8))) float v8f;

__device__ __forceinline__ float sigmoidf_(float x) {
    return 1.0f / (1.0f + expf(-x));
}

// =====================================================================
// WMMA f32 GEMM:  C[M,N] = A[M,K] @ W[N,K]^T + bias[N]
// A row-major [M,K], W row-major [N,K], C row-major [M,N].
// Block = 256 threads (8 waves). Block tile 128(M) x 64(N).
// K staged through double-buffered LDS in chunks of 16 (4 wmma k-steps),
// software-pipelined: next chunk's global b128 loads are in flight while
// the current chunk's 16 WMMAs consume LDS fragments.
// Requires: N % 64 == 0, K % 16 == 0, M % 16 == 0 (tiles predicated on M).
// =====================================================================
#define SROW 20                 // LDS row stride in floats (pad 16 -> 20)
#define ASZ  (128 * SROW)       // one A buffer (floats)
#define BSZ  (64 * SROW)        // one B buffer (floats)

__global__ __launch_bounds__(256) void gemm_wmma_f32(
    const float* __restrict__ A, const float* __restrict__ W,
    const float* __restrict__ bias, float* __restrict__ C,
    int M, int N, int K)
{
    __shared__ __align__(16) float smem[2 * ASZ + 2 * BSZ];

    const int tid  = threadIdx.x;
    const int lane = tid & 31;
    const int wave = tid >> 5;
    const int hi   = lane >> 4;

    const int m0b = blockIdx.y * 128;          // block M origin
    const int n0  = blockIdx.x * 64;           // block N origin
    const int m0  = m0b + wave * 16;           // wave M origin
    const int col = n0 + (lane & 15);          // N index for this lane

    // ---- global load indexing for tile staging ----
    // A tile: 128 rows x 16 cols = 512 float4; thread handles e=tid and e=tid+256
    const int ar0 = tid >> 2,         aq0 = (tid & 3) * 4;
    const int ar1 = (tid + 256) >> 2, aq1 = aq0;                  // same (e&3)
    // B tile: 64 rows x 16 cols = 256 float4; one per thread
    const int br  = tid >> 2,         bq  = (tid & 3) * 4;

    int agr0 = m0b + ar0; if (agr0 >= M) agr0 = M - 1;            // clamp (ragged M)
    int agr1 = m0b + ar1; if (agr1 >= M) agr1 = M - 1;

    const float* Ap0 = A + (size_t)agr0 * K + aq0;
    const float* Ap1 = A + (size_t)agr1 * K + aq1;
    const float* Bp  = W + (size_t)(n0 + br) * K + bq;

    // LDS float offsets for staging stores (b128-aligned: 80*r + 16*q bytes)
    const int sA0 = ar0 * SROW + aq0;
    const int sA1 = ar1 * SROW + aq1;
    const int sB  = 2 * ASZ + br * SROW + bq;

    // LDS float offsets for fragment reads
    const int aBase = (wave * 16 + (lane & 15)) * SROW;
    const int bBase = 2 * ASZ + (lane & 15) * SROW;

    v8f acc[4];
    #pragma unroll
    for (int nt = 0; nt < 4; ++nt) {
        float bv = bias[col + nt * 16];
        #pragma unroll
        for (int i = 0; i < 8; ++i) acc[nt][i] = bv;
    }

    const int nk = K >> 4;

    // ---- prologue: stage chunk 0 ----
    float4 ra0 = *(const float4*)(Ap0);
    float4 ra1 = *(const float4*)(Ap1);
    float4 rb  = *(const float4*)(Bp);
    *(float4*)&smem[sA0] = ra0;
    *(float4*)&smem[sA1] = ra1;
    *(float4*)&smem[sB]  = rb;
    __syncthreads();

    for (int kc = 0; kc < nk; ++kc) {
        const int p = kc & 1;
        // prefetch next chunk (loads fly under the WMMAs below)
        if (kc + 1 < nk) {
            const int ko = (kc + 1) << 4;
            ra0 = *(const float4*)(Ap0 + ko);
            ra1 = *(const float4*)(Ap1 + ko);
            rb  = *(const float4*)(Bp  + ko);
        }
        // compute chunk kc from buffer p
        const int aOff = p * ASZ + aBase;
        const int bOff = p * BSZ + bBase;
        #pragma unroll
        for (int s = 0; s < 4; ++s) {
            const int c = 4 * s + 2 * hi;
            v2f a  = *(const v2f*)&smem[aOff + c];
            v2f b0 = *(const v2f*)&smem[bOff + 0 * 16 * SROW + c];
            v2f b1 = *(const v2f*)&smem[bOff + 1 * 16 * SROW + c];
            v2f b2 = *(const v2f*)&smem[bOff + 2 * 16 * SROW + c];
            v2f b3 = *(const v2f*)&smem[bOff + 3 * 16 * SROW + c];
            acc[0] = __builtin_amdgcn_wmma_f32_16x16x4_f32(false, a, false, b0, (short)0, acc[0], false, false);
            acc[1] = __builtin_amdgcn_wmma_f32_16x16x4_f32(false, a, false, b1, (short)0, acc[1], false, false);
            acc[2] = __builtin_amdgcn_wmma_f32_16x16x4_f32(false, a, false, b2, (short)0, acc[2], false, false);
            acc[3] = __builtin_amdgcn_wmma_f32_16x16x4_f32(false, a, false, b3, (short)0, acc[3], false, false);
        }
        if (kc + 1 < nk) {
            __syncthreads();                       // everyone done reading p^1 (iter kc-1)
            const int q = p ^ 1;
            *(float4*)&smem[q * ASZ + sA0] = ra0;
            *(float4*)&smem[q * ASZ + sA1] = ra1;
            *(float4*)&smem[q * BSZ + sB]  = rb;   // note: sB already includes 2*ASZ
            __syncthreads();                       // buffer q ready
        }
    }

    // ---- store (predicated on ragged M; wave-tile is all-or-nothing) ----
    if (m0 < M) {
        #pragma unroll
        for (int nt = 0; nt < 4; ++nt) {
            float* crow = C + (size_t)(m0 + 8 * hi) * N + col + nt * 16;
            #pragma unroll
            for (int v = 0; v < 8; ++v)
                crow[(size_t)v * N] = acc[nt][v];
        }
    }
}

// =====================================================================
// Per-row dots: sL[m] = Wg_t[m,:].alpha_left ; sR[m] = Wg_t[m,:].alpha_right
// One wave per row.
// =====================================================================
__global__ __launch_bounds__(256) void scores_kernel(
    const float* __restrict__ Wgt, const float* __restrict__ aL,
    const float* __restrict__ aR, float* __restrict__ sL, float* __restrict__ sR)
{
    const int lane = threadIdx.x & 31;
    const int row  = blockIdx.x * 8 + (threadIdx.x >> 5);
    if (row >= BB * TT) return;
    const float* p = Wgt + (size_t)row * DD;
    float l = 0.f, r = 0.f;
    for (int d = lane; d < DD; d += 32) {
        float v = p[d];
        l = fmaf(v, aL[d], l);
        r = fmaf(v, aR[d], r);
    }
    #pragma unroll
    for (int off = 16; off; off >>= 1) {
        l += __shfl_xor(l, off, 32);
        r += __shfl_xor(r, off, 32);
    }
    if (lane == 0) { sL[row] = l; sR[row] = r; }
}

// =====================================================================
// CLS row init: G_W[b,0,:] = cls_embed
// =====================================================================
__global__ void cls_kernel(const float* __restrict__ cls, float* __restrict__ G_W)
{
    int i = blockIdx.x * blockDim.x + threadIdx.x;
    if (i < BB * DD) {
        int b = i / DD, d = i % DD;
        G_W[(size_t)b * TP1 * DD + d] = cls[d];
    }
}

// =====================================================================
// GAT attention + residual; writes G_W[b, t+1, :].  One wave per token.
// =====================================================================
__global__ __launch_bounds__(256) void gat_kernel(
    const float* __restrict__ emb, const float* __restrict__ Wgt,
    const float* __restrict__ sL, const float* __restrict__ sR,
    const int* __restrict__ child_idx, const int* __restrict__ child_mask,
    float* __restrict__ G_W)
{
    const int lane = threadIdx.x & 31;
    const int tok  = blockIdx.x * 8 + (threadIdx.x >> 5);
    if (tok >= BB * TT) return;
    const int b = tok >> 10;          // tok / T
    const int t = tok & (TT - 1);

    const float ss = sL[tok];
    float sc[KK];
    int   idx[KK];
    int   msk = 0;
    #pragma unroll
    for (int k = 0; k < KK; ++k) {
        int im = child_mask[tok * KK + k];
        int ci = child_idx[tok * KK + k];
        idx[k] = b * TT + ci;
        float s = ss + sR[idx[k]];
        s = (s < 0.f) ? 0.2f * s : s;           // leaky_relu(0.2)
        sc[k] = im ? s : -1e30f;
        msk |= (im ? 1 : 0) << k;
    }

    float w[KK];
    float inv = 0.f;
    if (msk) {
        float mx = -3.4e38f;
        #pragma unroll
        for (int k = 0; k < KK; ++k) if ((msk >> k) & 1) mx = fmaxf(mx, sc[k]);
        float sum = 0.f;
        #pragma unroll
        for (int k = 0; k < KK; ++k) {
            float e = ((msk >> k) & 1) ? expf(sc[k] - mx) : 0.f;
            w[k] = e; sum += e;
        }
        inv = 1.f / sum;
    }

    // acc = emb row ; then += a_k * Wg_t[child_k]
    const float4* e4 = (const float4*)(emb + (size_t)tok * DD);
    float4 acc[6];
    #pragma unroll
    for (int j = 0; j < 6; ++j) acc[j] = e4[lane + 32 * j];

    if (msk) {
        #pragma unroll
        for (int k = 0; k < KK; ++k) {
            if (!((msk >> k) & 1)) continue;
            const float ak = w[k] * inv;
            const float4* c4 = (const float4*)(Wgt + (size_t)idx[k] * DD);
            #pragma unroll
            for (int j = 0; j < 6; ++j) {
                float4 cv = c4[lane + 32 * j];
                acc[j].x = fmaf(ak, cv.x, acc[j].x);
                acc[j].y = fmaf(ak, cv.y, acc[j].y);
                acc[j].z = fmaf(ak, cv.z, acc[j].z);
                acc[j].w = fmaf(ak, cv.w, acc[j].w);
            }
        }
    }

    float4* o4 = (float4*)(G_W + ((size_t)b * TP1 + t + 1) * DD);
    #pragma unroll
    for (int j = 0; j < 6; ++j) o4[lane + 32 * j] = acc[j];
}

// =====================================================================
// GRU scan. One block per batch row (independent scans -> no grid sync).
// blockDim = 576: thread tid computes hp[tid] and hp[tid+576].
// Steps with mask==0 are skipped entirely (h unchanged).
// =====================================================================
__global__ __launch_bounds__(576) void gru_kernel(
    const float* __restrict__ xp,        // [B, T+1, 3H]
    const int* __restrict__ clue_mask,   // [B, T]
    const float* __restrict__ Whh,       // [3H, H]
    const float* __restrict__ bhh,       // [3H]
    float* __restrict__ h_out)           // [B, H]
{
    __shared__ __align__(16) float h[HH];
    __shared__ float hp[TH3];
    const int b   = blockIdx.x;
    const int tid = threadIdx.x;

    if (tid < HH) h[tid] = 0.f;
    __syncthreads();

    const float* xpb = xp + (size_t)b * TP1 * TH3;
    const int n0 = tid, n1 = tid + 576;
    const float4* w0 = (const float4*)(Whh + (size_t)n0 * HH);
    const float4* w1 = (const float4*)(Whh + (size_t)n1 * HH);
    const float4* h4 = (const float4*)h;

    for (int t = 0; t < TP1; ++t) {
        const int m = (t == 0) ? 1 : clue_mask[b * TT + (t - 1)];
        if (m == 0) continue;                   // block-uniform skip

        // two interleaved partial sums per output -> half the FMA chain depth
        float s0a = bhh[n0], s1a = bhh[n1], s0b = 0.f, s1b = 0.f;
        #pragma unroll 4
        for (int j = 0; j < HH / 4; j += 2) {
            float4 hva = h4[j],     hvb = h4[j + 1];
            float4 aa  = w0[j],     ab  = w0[j + 1];
            float4 ca  = w1[j],     cb  = w1[j + 1];
            s0a = fmaf(aa.x, hva.x, fmaf(aa.y, hva.y, fmaf(aa.z, hva.z, fmaf(aa.w, hva.w, s0a))));
            s0b = fmaf(ab.x, hvb.x, fmaf(ab.y, hvb.y, fmaf(ab.z, hvb.z, fmaf(ab.w, hvb.w, s0b))));
            s1a = fmaf(ca.x, hva.x, fmaf(ca.y, hva.y, fmaf(ca.z, hva.z, fmaf(ca.w, hva.w, s1a))));
            s1b = fmaf(cb.x, hvb.x, fmaf(cb.y, hvb.y, fmaf(cb.z, hvb.z, fmaf(cb.w, hvb.w, s1b))));
        }
        hp[n0] = s0a + s0b;
        hp[n1] = s1a + s1b;
        __syncthreads();                        // hp ready; all h reads done

        if (tid < HH) {
            const float* xrow = xpb + (size_t)t * TH3;
            float r  = sigmoidf_(xrow[tid]         + hp[tid]);
            float z  = sigmoidf_(xrow[HH + tid]    + hp[HH + tid]);
            float ng = tanhf(xrow[2 * HH + tid] + r * hp[2 * HH + tid]);
            h[tid] = (1.f - z) * ng + z * h[tid];
        }
        __syncthreads();                        // h updated before next step reads
    }

    if (tid < HH) h_out[b * HH + tid] = h[tid];
}

// =====================================================================
// Output heads: per token, 6 dots of length 1152 over [G_W row | h_clue[b]].
// One wave per token. out = [O_cause (B*T*3) | O_effect (B*T*3)].
// =====================================================================
__global__ __launch_bounds__(256) void heads_kernel(
    const float* __restrict__ G_W, const float* __restrict__ hcl,
    const float* __restrict__ Wc_w, const float* __restrict__ Wc_b,
    const float* __restrict__ We_w, const float* __restrict__ We_b,
    float* __restrict__ out)
{
    const int lane = threadIdx.x & 31;
    const int tok  = blockIdx.x * 8 + (threadIdx.x >> 5);
    if (tok >= BB * TT) return;
    const int b = tok >> 10;
    const int t = tok & (TT - 1);

    const float* grow = G_W + ((size_t)b * TP1 + t + 1) * DD;
    const float* hrow = hcl + b * HH;

    float a[6] = {0.f, 0.f, 0.f, 0.f, 0.f, 0.f};
    for (int d = lane; d < TH3; d += 32) {      // d<DD boundary is wave-uniform
        float f = (d < DD) ? grow[d] : hrow[d - DD];
        a[0] = fmaf(f, Wc_w[0 * TH3 + d], a[0]);
        a[1] = fmaf(f, Wc_w[1 * TH3 + d], a[1]);
        a[2] = fmaf(f, Wc_w[2 * TH3 + d], a[2]);
        a[3] = fmaf(f, We_w[0 * TH3 + d], a[3]);
        a[4] = fmaf(f, We_w[1 * TH3 + d], a[4]);
        a[5] = fmaf(f, We_w[2 * TH3 + d], a[5]);
    }
    #pragma unroll
    for (int off = 16; off; off >>= 1) {
        #pragma unroll
        for (int q = 0; q < 6; ++q) a[q] += __shfl_xor(a[q], off, 32);
    }
    if (lane == 0) {
        const size_t o = (size_t)tok * 3;
        out[o + 0] = a[0] + Wc_b[0];
        out[o + 1] = a[1] + Wc_b[1];
        out[o + 2] = a[2] + Wc_b[2];
        out[(size_t)BB * TT * 3 + o + 0] = a[3] + We_b[0];
        out[(size_t)BB * TT * 3 + o + 1] = a[4] + We_b[1];
        out[(size_t)BB * TT * 3 + o + 2] = a[5] + We_b[2];
    }
}

extern "C" void kernel_launch(void* const* d_in, const int* in_sizes, int n_in,
                              void* d_out, int out_size, void* d_ws, size_t ws_size,
                              hipStream_t stream) {
    (void)in_sizes; (void)n_in; (void)out_size; (void)ws_size;

    const float* emb       = (const float*)d_in[0];
    const float* cls_embed = (const float*)d_in[1];
    const float* Wg_w      = (const float*)d_in[2];
    const float* Wg_b      = (const float*)d_in[3];
    const float* alpha_l   = (const float*)d_in[4];
    const float* alpha_r   = (const float*)d_in[5];
    const float* gru_Wih   = (const float*)d_in[6];
    const float* gru_bih   = (const float*)d_in[7];
    const float* gru_Whh   = (const float*)d_in[8];
    const float* gru_bhh   = (const float*)d_in[9];
    const float* Wc_w      = (const float*)d_in[10];
    const float* Wc_b      = (const float*)d_in[11];
    const float* We_w      = (const float*)d_in[12];
    const float* We_b      = (const float*)d_in[13];
    const int*   child_idx = (const int*)d_in[14];
    const int*   child_msk = (const int*)d_in[15];
    const int*   clue_mask = (const int*)d_in[16];
    float* out = (float*)d_out;

    // Workspace layout (floats):
    //  bufA : 16400*1152  (holds Wg_t first, reused for x_proj)
    //  G_W  : 16400*768
    //  sL,sR: 16384 each ; hcl: 16*384
    float* ws   = (float*)d_ws;
    float* bufA = ws;
    float* G_W  = bufA + (size_t)16400 * TH3;
    float* sL   = G_W  + (size_t)16400 * DD;
    float* sR   = sL + BB * TT;
    float* hcl  = sR + BB * TT;

    // 1) Wg_t = emb @ Wg_w^T + Wg_b        [16384 x 768]
    {
        dim3 grid(DD / 64, (BB * TT + 127) / 128);
        gemm_wmma_f32<<<grid, 256, 0, stream>>>(emb, Wg_w, Wg_b, bufA,
                                                BB * TT, DD, DD);
    }
    // 2) score vectors sL, sR
    scores_kernel<<<(BB * TT + 7) / 8, 256, 0, stream>>>(bufA, alpha_l, alpha_r, sL, sR);
    // 3) CLS rows of G_W
    cls_kernel<<<(BB * DD + 255) / 256, 256, 0, stream>>>(cls_embed, G_W);
    // 4) GAT + residual -> G_W[:,1:,:]
    gat_kernel<<<(BB * TT + 7) / 8, 256, 0, stream>>>(emb, bufA, sL, sR,
                                                      child_idx, child_msk, G_W);
    // 5) x_proj = G_W @ gru_Wih^T + gru_bih   [16400 x 1152]  (reuses bufA)
    {
        dim3 grid(TH3 / 64, (BB * TP1 + 127) / 128);
        gemm_wmma_f32<<<grid, 256, 0, stream>>>(G_W, gru_Wih, gru_bih, bufA,
                                                BB * TP1, TH3, DD);
    }
    // 6) GRU scan (one WGP per batch row)
    gru_kernel<<<BB, 576, 0, stream>>>(bufA, clue_mask, gru_Whh, gru_bhh, hcl);
    // 7) cause / effect heads
    heads_kernel<<<(BB * TT + 7) / 8, 256, 0, stream>>>(G_W, hcl, Wc_w, Wc_b,
                                                        We_w, We_b, out);
}